// GaussianMaskRenderer_89618787598722
// MI455X (gfx1250) — compile-verified
//
#include <hip/hip_runtime.h>
#include <math.h>

// ---------------- problem constants ----------------
constexpr int Hh = 224, Ww = 448;
constexpr int Tt = 2, Vv = 2, Mv = 4;           // M = B*T*V views
constexpr int Np = 12544;                        // V*GH*GW gaussians
constexpr int NP2 = 16384;                       // sort padding (pow2)
constexpr int CHUNK = 256;                       // points per raster chunk (Np = 49*256)
constexpr int NCHUNK = Np / CHUNK;               // 49

constexpr long long PLANE = (long long)Hh * Ww;  // 100352
// output float offsets (return order of the reference)
constexpr long long RGB_OFS[3] = { 0LL, 1204224LL, 2408448LL };        // sta, dyn, all
constexpr long long AL_OFS[3]  = { 3612672LL, 4014080LL, 4415488LL };  // sta, dyn, all
constexpr long long SEM_OFS    = 4816896LL;
constexpr long long SIG_OFS    = 6021120LL;
constexpr long long TOUCH_OFS  = 6021121LL;
constexpr long long SEM_CNT    = 1204224LL;

// workspace float offsets
constexpr int WS_MATS  = 0;                      // 4 * 16 combined matrices
constexpr int WS_STATS = 64;                     // [0..3]=sigma sums, [4..7]=counts, [8]=touch
constexpr int WS_ZKEY  = 80;                     // Mv*Np floats
constexpr int WS_SIDX  = WS_ZKEY + Mv * Np;      // Mv*Np ints
constexpr int WS_Q0    = WS_SIDX + Mv * Np;      // Mv*Np float4 (u,v,sigma,a_sta)
constexpr int WS_Q1    = WS_Q0 + 4 * Mv * Np;    // Mv*Np float4 (a_dyn,a_all,r,g)
constexpr int WS_Q2    = WS_Q1 + 4 * Mv * Np;    // Mv*Np float4 (b,zkey,0,0)

typedef float v2f __attribute__((ext_vector_type(2)));
typedef float v8f __attribute__((ext_vector_type(8)));

__device__ __forceinline__ float clamp01(float x) { return fminf(fmaxf(x, 0.0f), 1.0f); }

// ---------------- 0) zero the stat counters ----------------
__global__ void init_stats_kernel(float* __restrict__ ws) {
    if (threadIdx.x < 16) ws[WS_STATS + threadIdx.x] = 0.0f;
}

// ---------------- 1) per-view combined matrix: inv(c2w) @ first_pose ----------------
__global__ void prep_mats_kernel(const float* __restrict__ c2w,
                                 const float* __restrict__ fpose,
                                 float* __restrict__ ws) {
    int m = threadIdx.x;
    if (m >= Mv) return;
    const float* a = c2w + m * 16;
    float inv[16];
    inv[0]  =  a[5]*a[10]*a[15] - a[5]*a[11]*a[14] - a[9]*a[6]*a[15] + a[9]*a[7]*a[14] + a[13]*a[6]*a[11] - a[13]*a[7]*a[10];
    inv[4]  = -a[4]*a[10]*a[15] + a[4]*a[11]*a[14] + a[8]*a[6]*a[15] - a[8]*a[7]*a[14] - a[12]*a[6]*a[11] + a[12]*a[7]*a[10];
    inv[8]  =  a[4]*a[9]*a[15]  - a[4]*a[11]*a[13] - a[8]*a[5]*a[15] + a[8]*a[7]*a[13] + a[12]*a[5]*a[11] - a[12]*a[7]*a[9];
    inv[12] = -a[4]*a[9]*a[14]  + a[4]*a[10]*a[13] + a[8]*a[5]*a[14] - a[8]*a[6]*a[13] - a[12]*a[5]*a[10] + a[12]*a[6]*a[9];
    inv[1]  = -a[1]*a[10]*a[15] + a[1]*a[11]*a[14] + a[9]*a[2]*a[15] - a[9]*a[3]*a[14] - a[13]*a[2]*a[11] + a[13]*a[3]*a[10];
    inv[5]  =  a[0]*a[10]*a[15] - a[0]*a[11]*a[14] - a[8]*a[2]*a[15] + a[8]*a[3]*a[14] + a[12]*a[2]*a[11] - a[12]*a[3]*a[10];
    inv[9]  = -a[0]*a[9]*a[15]  + a[0]*a[11]*a[13] + a[8]*a[1]*a[15] - a[8]*a[3]*a[13] - a[12]*a[1]*a[11] + a[12]*a[3]*a[9];
    inv[13] =  a[0]*a[9]*a[14]  - a[0]*a[10]*a[13] - a[8]*a[1]*a[14] + a[8]*a[2]*a[13] + a[12]*a[1]*a[10] - a[12]*a[2]*a[9];
    inv[2]  =  a[1]*a[6]*a[15]  - a[1]*a[7]*a[14]  - a[5]*a[2]*a[15] + a[5]*a[3]*a[14] + a[13]*a[2]*a[7]  - a[13]*a[3]*a[6];
    inv[6]  = -a[0]*a[6]*a[15]  + a[0]*a[7]*a[14]  + a[4]*a[2]*a[15] - a[4]*a[3]*a[14] - a[12]*a[2]*a[7]  + a[12]*a[3]*a[6];
    inv[10] =  a[0]*a[5]*a[15]  - a[0]*a[7]*a[13]  - a[4]*a[1]*a[15] + a[4]*a[3]*a[13] + a[12]*a[1]*a[7]  - a[12]*a[3]*a[5];
    inv[14] = -a[0]*a[5]*a[14]  + a[0]*a[6]*a[13]  + a[4]*a[1]*a[14] - a[4]*a[2]*a[13] - a[12]*a[1]*a[6]  + a[12]*a[2]*a[5];
    inv[3]  = -a[1]*a[6]*a[11]  + a[1]*a[7]*a[10]  + a[5]*a[2]*a[11] - a[5]*a[3]*a[10] - a[9]*a[2]*a[7]   + a[9]*a[3]*a[6];
    inv[7]  =  a[0]*a[6]*a[11]  - a[0]*a[7]*a[10]  - a[4]*a[2]*a[11] + a[4]*a[3]*a[10] + a[8]*a[2]*a[7]   - a[8]*a[3]*a[6];
    inv[11] = -a[0]*a[5]*a[11]  + a[0]*a[7]*a[9]   + a[4]*a[1]*a[11] - a[4]*a[3]*a[9]  - a[8]*a[1]*a[7]   + a[8]*a[3]*a[5];
    inv[15] =  a[0]*a[5]*a[10]  - a[0]*a[6]*a[9]   - a[4]*a[1]*a[10] + a[4]*a[2]*a[9]  + a[8]*a[1]*a[6]   - a[8]*a[2]*a[5];
    float det = a[0]*inv[0] + a[1]*inv[4] + a[2]*inv[8] + a[3]*inv[12];
    float rdet = 1.0f / det;
    float* mc = ws + WS_MATS + m * 16;
    // mc = inv(c2w) @ first_pose
    for (int i = 0; i < 4; ++i)
        for (int j = 0; j < 4; ++j) {
            float s = 0.0f;
            for (int k = 0; k < 4; ++k) s += inv[i*4+k] * rdet * fpose[k*4+j];
            mc[i*4+j] = s;
        }
}

// ---------------- 2) projection: WMMA 16x16x4 point transform + attribute prep ----------------
__global__ __launch_bounds__(256)
void project_kernel(const float* __restrict__ center, const float* __restrict__ scal,
                    const float* __restrict__ feat,   const float* __restrict__ opac,
                    const float* __restrict__ bgp,    const float* __restrict__ intr,
                    float* __restrict__ ws) {
    const int m    = blockIdx.y;            // view index 0..3 (= t*V + vc)
    const int t    = m / Vv;
    const int vc   = m % Vv;
    const int tid  = threadIdx.x;
    const int wave = tid >> 5;
    const int lane = tid & 31;
    const int n0   = blockIdx.x * 256 + wave * 32;  // first point of this wave

    __shared__ float camLds[8][32][3];

    const float* mc = ws + WS_MATS + m * 16;   // combined 4x4, row major

    // B matrix (4x16): B[k][n] = mc[n][k]; lanes 0-15 hold K={0,1}, lanes 16-31 K={2,3}
    // Unconditional loads + VALU select: no exec-mask branches feeding the WMMAs.
    const int ncol = lane & 15;
    const int kb   = (lane >> 4) * 2;
    const int ci   = (ncol < 4) ? ncol : 0;
    const float be0 = mc[ci * 4 + kb];
    const float be1 = mc[ci * 4 + kb + 1];
    v2f bm;
    bm.x = (ncol < 4) ? be0 : 0.0f;
    bm.y = (ncol < 4) ? be1 : 0.0f;

    // A matrices (16x4): row = lane&15, lanes 0-15 carry (x,y), lanes 16-31 carry (z,1)
    const bool hiK = lane >= 16;
    const int  p1  = t * Np + n0 + (lane & 15);
    const int  p2  = p1 + 16;
    const float* c1 = center + (long long)p1 * 3;
    const float* c2 = center + (long long)p2 * 3;
    const float x1 = c1[0], y1 = c1[1], z1 = c1[2];
    const float x2 = c2[0], y2 = c2[1], z2 = c2[2];
    v2f a1, a2;
    a1.x = hiK ? z1 : x1;
    a1.y = hiK ? 1.0f : y1;
    a2.x = hiK ? z2 : x2;
    a2.y = hiK ? 1.0f : y2;

    v8f cz = {};
    v8f d1 = __builtin_amdgcn_wmma_f32_16x16x4_f32(false, a1, false, bm, (short)0, cz, false, false);
    v8f d2 = __builtin_amdgcn_wmma_f32_16x16x4_f32(false, a2, false, bm, (short)0, cz, false, false);

    // D[mrow][n]: lane holds column n = lane&15 for rows r + 8*(lane>>4); keep n<3 (cam x,y,z)
    if (ncol < 3) {
        const int mb = 8 * (lane >> 4);
        #pragma unroll
        for (int r = 0; r < 8; ++r) {
            camLds[wave][mb + r][ncol]      = d1[r];
            camLds[wave][16 + mb + r][ncol] = d2[r];
        }
    }
    __syncthreads();

    // per-lane point math
    const int n  = n0 + lane;
    const int pt = t * Np + n;
    const float cxx = camLds[wave][lane][0];
    const float cyy = camLds[wave][lane][1];
    const float czz = camLds[wave][lane][2];

    const float fx = intr[vc * 4 + 0], fy = intr[vc * 4 + 1];
    const float cx0 = intr[vc * 4 + 2], cy0 = intr[vc * 4 + 3];

    const float baseA = clamp01(opac[pt]);
    const float dynp  = clamp01(1.0f - bgp[pt]);
    const float aAll  = baseA;
    const float aDyn  = baseA * dynp;
    const float aSta  = baseA * (1.0f - dynp);

    const bool finite = isfinite(cxx) && isfinite(cyy) && isfinite(czz);
    const bool valid  = (czz > 0.001f) && finite && (aAll > 1e-5f);
    const float zsafe = valid ? czz : 1.0f;
    const float u = cxx * fx / zsafe + cx0;
    const float v = cyy * fy / zsafe + cy0;
    const float sm = (scal[pt*3+0] + scal[pt*3+1] + scal[pt*3+2]) * (1.0f / 3.0f);
    const float sig = fminf(fmaxf((fx + fy) * 0.5f * fabsf(sm) / fmaxf(zsafe, 0.001f), 0.75f), 10.0f);
    const bool inb = (u >= -3.0f) && (u <= (float)Ww + 2.0f) && (v >= -3.0f) && (v <= (float)Hh + 2.0f);
    const bool keep = valid && inb;

    if (keep) {
        atomicAdd(ws + WS_STATS + m, sig);
        atomicAdd(ws + WS_STATS + 4 + m, 1.0f);
    }

    const float uS = keep ? u : -1.0e6f;
    const float vS = keep ? v : -1.0e6f;
    const float zk = keep ? czz : INFINITY;
    const float cr = clamp01(feat[pt*3+0]);
    const float cg = clamp01(feat[pt*3+1]);
    const float cb = clamp01(feat[pt*3+2]);
    const float aS = (keep && aSta > 1e-5f) ? aSta : 0.0f;
    const float aD = (keep && aDyn > 1e-5f) ? aDyn : 0.0f;
    const float aA = keep ? aAll : 0.0f;

    const int gi = m * Np + n;
    ws[WS_ZKEY + gi] = zk;
    ((float4*)(ws + WS_Q0))[gi] = make_float4(uS, vS, sig, aS);
    ((float4*)(ws + WS_Q1))[gi] = make_float4(aD, aA, cr, cg);
    ((float4*)(ws + WS_Q2))[gi] = make_float4(cb, zk, 0.0f, 0.0f);
}

// ---------------- 3) per-view depth sort: 128KB LDS bitonic (CDNA5 320KB LDS) ----------------
__global__ __launch_bounds__(1024)
void sort_kernel(float* __restrict__ ws) {
    const int m = blockIdx.x;
    extern __shared__ char smem[];
    float* key = (float*)smem;
    int*   idx = (int*)(smem + NP2 * sizeof(float));
    const int tid = threadIdx.x;

    const float* zkey = ws + WS_ZKEY + m * Np;
    for (int i = tid; i < NP2; i += 1024) {
        key[i] = (i < Np) ? zkey[i] : INFINITY;
        idx[i] = i;
    }
    __syncthreads();

    for (int k = 2; k <= NP2; k <<= 1) {
        for (int j = k >> 1; j > 0; j >>= 1) {
            for (int i = tid; i < NP2; i += 1024) {
                const int ixj = i ^ j;
                if (ixj > i) {
                    const bool up = ((i & k) == 0);
                    const float ki = key[i], kj = key[ixj];
                    if (up ? (ki > kj) : (ki < kj)) {
                        key[i] = kj; key[ixj] = ki;
                        const int tsw = idx[i]; idx[i] = idx[ixj]; idx[ixj] = tsw;
                    }
                }
            }
            __syncthreads();
        }
    }
    int* sidx = ((int*)(ws + WS_SIDX)) + m * Np;
    for (int i = tid; i < Np; i += 1024) sidx[i] = idx[i];
}

// ---------------- 4) tile raster: async-LDS double-buffered 3-branch compositing ----------------
__global__ __launch_bounds__(256)
void raster_kernel(const float* __restrict__ ws, float* __restrict__ out) {
    const int m  = blockIdx.z;
    const int px = blockIdx.x * 16 + threadIdx.x;
    const int py = blockIdx.y * 16 + threadIdx.y;
    const int tid = threadIdx.y * 16 + threadIdx.x;

    __shared__ float4 s0[2][CHUNK], s1[2][CHUNK], s2[2][CHUNK];

    const float4* q0 = ((const float4*)(ws + WS_Q0)) + m * Np;
    const float4* q1 = ((const float4*)(ws + WS_Q1)) + m * Np;
    const float4* q2 = ((const float4*)(ws + WS_Q2)) + m * Np;
    const int*  sidx = ((const int*)(ws + WS_SIDX)) + m * Np;

    // Issue 3 async global->LDS B128 gathers for chunk c into buffer `buf`.
    // ASYNCcnt is per-wave and completes in order: waiting for <=3 outstanding
    // guarantees the previous chunk's 3 transfers have landed in LDS.
    auto issueChunk = [&](int c, int buf) -> int {
        const int sid = sidx[c * CHUNK + tid];
        const int sc  = (sid < Np) ? sid : 0;
        const unsigned long long a0 = (unsigned long long)(const void*)(q0 + sc);
        const unsigned long long a1 = (unsigned long long)(const void*)(q1 + sc);
        const unsigned long long a2 = (unsigned long long)(const void*)(q2 + sc);
        const unsigned l0 = (unsigned)(unsigned long long)(const void*)&s0[buf][tid];
        const unsigned l1 = (unsigned)(unsigned long long)(const void*)&s1[buf][tid];
        const unsigned l2 = (unsigned)(unsigned long long)(const void*)&s2[buf][tid];
        asm volatile("global_load_async_to_lds_b128 %0, %1, off" :: "v"(l0), "v"(a0) : "memory");
        asm volatile("global_load_async_to_lds_b128 %0, %1, off" :: "v"(l1), "v"(a1) : "memory");
        asm volatile("global_load_async_to_lds_b128 %0, %1, off" :: "v"(l2), "v"(a2) : "memory");
        return sid;
    };

    float aAcc[3] = {0.0f, 0.0f, 0.0f};
    float cAcc[3][3] = {{0,0,0},{0,0,0},{0,0,0}};
    const float fpx = (float)px, fpy = (float)py;

    int sidCur = issueChunk(0, 0);
    for (int c = 0; c < NCHUNK; ++c) {
        const int buf = c & 1;
        int sidNext = 0;
        if (c + 1 < NCHUNK) {
            sidNext = issueChunk(c + 1, (c + 1) & 1);
            if (c + 2 < NCHUNK) __builtin_prefetch(sidx + (c + 2) * CHUNK + tid, 0, 1);
            asm volatile("s_wait_asynccnt 0x3" ::: "memory");   // chunk c resident
        } else {
            asm volatile("s_wait_asynccnt 0x0" ::: "memory");
        }
        if (sidCur >= Np) s0[buf][tid].x = -1.0e6f;   // padded sort slot -> dead point
        __syncthreads();

        for (int j = 0; j < CHUNK; ++j) {
            const float u  = s0[buf][j].x;
            const float v  = s0[buf][j].y;
            const float sg = s0[buf][j].z;
            const int x0 = (int)floorf(u);
            const int y0 = (int)floorf(v);
            int rad = (int)ceilf(sg * 1.5f);
            rad = rad < 1 ? 1 : (rad > 2 ? 2 : rad);
            const int dx = px - x0, dy = py - y0;
            if (dx < -rad || dx > rad || dy < -rad || dy > rad) continue;

            const float scv = fmaxf(sg, 0.001f);
            const float du = (u - fpx) / scv;
            const float dv = (v - fpy) / scv;
            const float g = __expf(-0.5f * (du * du + dv * dv));

            const float av[3] = { s0[buf][j].w, s1[buf][j].x, s1[buf][j].y };   // sta, dyn, all
            const float col[3] = { s1[buf][j].z, s1[buf][j].w, s2[buf][j].x };  // r, g, b
            #pragma unroll
            for (int b = 0; b < 3; ++b) {
                const float la = fminf(g * av[b], 0.999f);
                const float tr = 1.0f - aAcc[b];                      // aAcc in [0,0.999]
                const float contrib = la * tr;
                cAcc[b][0] += col[0] * contrib;
                cAcc[b][1] += col[1] * contrib;
                cAcc[b][2] += col[2] * contrib;
                aAcc[b] = fminf(aAcc[b] + contrib, 0.999f);
            }
        }
        __syncthreads();
        sidCur = sidNext;
    }

    const long long pix = (long long)py * Ww + px;
    #pragma unroll
    for (int b = 0; b < 3; ++b) {
        #pragma unroll
        for (int ch = 0; ch < 3; ++ch)
            out[RGB_OFS[b] + ((long long)(m * 3 + ch)) * PLANE + pix] = clamp01(cAcc[b][ch]);
        out[AL_OFS[b] + (long long)m * PLANE + pix] = aAcc[b];
    }
}

// ---------------- 5) touch count over the "all" alpha plane ----------------
__global__ __launch_bounds__(256)
void touch_kernel(const float* __restrict__ out, float* __restrict__ ws) {
    __shared__ float red[256];
    const long long total = (long long)Mv * PLANE;
    const long long i = (long long)blockIdx.x * 256 + threadIdx.x;
    float vv = 0.0f;
    if (i < total) vv = (out[AL_OFS[2] + i] > 1e-6f) ? 1.0f : 0.0f;
    red[threadIdx.x] = vv;
    __syncthreads();
    for (int s = 128; s > 0; s >>= 1) {
        if (threadIdx.x < s) red[threadIdx.x] += red[threadIdx.x + s];
        __syncthreads();
    }
    if (threadIdx.x == 0) atomicAdd(ws + WS_STATS + 8, red[0]);
}

// ---------------- 6) scalars ----------------
__global__ void finalize_kernel(const float* __restrict__ ws, float* __restrict__ out) {
    if (threadIdx.x != 0 || blockIdx.x != 0) return;
    float sm = 0.0f;
    for (int m = 0; m < Mv; ++m) {
        const float s = ws[WS_STATS + m];
        const float c = ws[WS_STATS + 4 + m];
        sm += (c > 0.0f) ? s / fmaxf(c, 1.0f) : 0.0f;
    }
    out[SIG_OFS]   = sm / (float)Mv;
    out[TOUCH_OFS] = ws[WS_STATS + 8] / (float)((long long)Mv * PLANE);
}

extern "C" void kernel_launch(void* const* d_in, const int* in_sizes, int n_in,
                              void* d_out, int out_size, void* d_ws, size_t ws_size,
                              hipStream_t stream) {
    const float* center = (const float*)d_in[0];
    const float* scal   = (const float*)d_in[1];
    const float* feat   = (const float*)d_in[2];
    const float* opac   = (const float*)d_in[3];
    const float* bgp    = (const float*)d_in[4];
    const float* sem    = (const float*)d_in[5];
    const float* intr   = (const float*)d_in[6];
    const float* c2w    = (const float*)d_in[7];
    const float* fpose  = (const float*)d_in[8];
    float* out = (float*)d_out;
    float* ws  = (float*)d_ws;

    // pass-through tensor
    hipMemcpyAsync(out + SEM_OFS, sem, (size_t)SEM_CNT * sizeof(float),
                   hipMemcpyDeviceToDevice, stream);

    init_stats_kernel<<<1, 32, 0, stream>>>(ws);
    prep_mats_kernel<<<1, 32, 0, stream>>>(c2w, fpose, ws);
    project_kernel<<<dim3(Np / 256, Mv), 256, 0, stream>>>(center, scal, feat, opac, bgp, intr, ws);
    sort_kernel<<<Mv, 1024, NP2 * (sizeof(float) + sizeof(int)), stream>>>(ws);
    raster_kernel<<<dim3(Ww / 16, Hh / 16, Mv), dim3(16, 16), 0, stream>>>(ws, out);
    touch_kernel<<<(int)((Mv * PLANE + 255) / 256), 256, 0, stream>>>(out, ws);
    finalize_kernel<<<1, 1, 0, stream>>>(ws, out);
}